// EcTransOctahedralSHTModule_88098369175964
// MI455X (gfx1250) — compile-verified
//
#include <hip/hip_runtime.h>

typedef float v2f __attribute__((ext_vector_type(2)));
typedef float v8f __attribute__((ext_vector_type(8)));

#define NLAT_    160
#define NRINGS_  320
#define MAXLON_  656
#define MMAX_    160
#define NVAR_    80
#define NB_      2
#define NPOINTS_ 108160
#define T_SPEC_  159

__device__ __forceinline__ int nlon_of(int r) {
  int i = (r < NLAT_) ? r : (NRINGS_ - 1 - r);
  return 20 + 4 * i;
}
__device__ __forceinline__ int cum_of(int r) {
  if (r < NLAT_) return 2 * r * r + 18 * r;
  int q = NRINGS_ - r;
  return NPOINTS_ - (2 * q * q + 18 * q);
}

// ---------------------------------------------------------------------------
// Stage 1: one block per ring. F[b][ri][m][v] = sum_n W_ri[r][m][n]*X[b][n][v]
// for BOTH batches (W streamed once, A-fragment reused 10x: 2 b x 5 v-tiles).
// 640 threads = 20 waves: wave = (ri in 0..1) x (mtile in 0..9).
// X panel is double-buffered through LDS with CDNA5 async-to-LDS copies:
// each chunk = 32 k-rows x 80 v x 2 b = exactly 2 async b128 per thread
// (rows clamped to NLON-1 so EXEC is always full and the per-wave ASYNCcnt
// per batch is exactly 2; async loads retire in order -> s_wait_asynccnt 2
// guarantees the previous batch landed).
// ---------------------------------------------------------------------------
#define S1_KCH 32   // k-rows per chunk; LDS = 2 buf * 2 b * 32 * 80 * 4 = 40 KB

__global__ __launch_bounds__(640) void sht_dft_stage1(
    const float* __restrict__ x, const float* __restrict__ Wr,
    const float* __restrict__ Wi, float* __restrict__ F)
{
  __shared__ float Xs[2][2][S1_KCH][NVAR_];   // [buf][b][row][v]

  const int r    = blockIdx.x;
  const int nlon = nlon_of(r);
  const int cum  = cum_of(r);
  const int hzw  = min(T_SPEC_, nlon >> 1);
  const int mt_used = (hzw + 1 + 15) >> 4;

  const int tid  = threadIdx.x;
  const int wave = tid >> 5;
  const int lane = tid & 31;
  const int lrow = lane & 15;
  const int lhi  = lane >> 4;

  const int ri    = wave / 10;   // 0 = real, 1 = imag
  const int mtile = wave % 10;
  const int m0    = mtile * 16;
  const bool active = (mtile < mt_used);

  const float* Wbase = (ri == 0 ? Wr : Wi) +
      ((size_t)r * MMAX_ + (size_t)(m0 + lrow)) * MAXLON_ + 2 * lhi;

  // decompose copy-unit id once: u = tid + it*640 over (b2, n, f4)
  const int f0  = tid % (NVAR_ / 4);
  const int n0  = (tid / (NVAR_ / 4)) % S1_KCH;
  const int b20 = tid / ((NVAR_ / 4) * S1_KCH);          // it=0 slice
  const int u1  = tid + 640;
  const int f1  = u1 % (NVAR_ / 4);
  const int n1  = (u1 / (NVAR_ / 4)) % S1_KCH;
  const int b21 = u1 / ((NVAR_ / 4) * S1_KCH);           // it=1 slice

  const unsigned lds0 = (unsigned)(size_t)(&Xs[0][0][0][0]);

  v8f acc[2][5] = {};   // [b][v-tile]

  const int nchunks = (nlon + S1_KCH - 1) / S1_KCH;

  // --- issue helper (manually unrolled twice for uniform ASYNCcnt) ---------
  auto issue = [&](int kc, int buf) {
    {
      const int ng = min(kc + n0, nlon - 1);
      const unsigned voff =
          (unsigned)(((b20 * NPOINTS_ + cum + ng) * NVAR_ + f0 * 4) * 4);
      const unsigned ldst = lds0 +
          (unsigned)((((buf * 2 + b20) * S1_KCH + n0) * NVAR_ + f0 * 4) * 4);
      asm volatile("global_load_async_to_lds_b128 %0, %1, %2"
                   :: "v"(ldst), "v"(voff), "s"(x) : "memory");
    }
    {
      const int ng = min(kc + n1, nlon - 1);
      const unsigned voff =
          (unsigned)(((b21 * NPOINTS_ + cum + ng) * NVAR_ + f1 * 4) * 4);
      const unsigned ldst = lds0 +
          (unsigned)((((buf * 2 + b21) * S1_KCH + n1) * NVAR_ + f1 * 4) * 4);
      asm volatile("global_load_async_to_lds_b128 %0, %1, %2"
                   :: "v"(ldst), "v"(voff), "s"(x) : "memory");
    }
  };

  issue(0, 0);

  for (int c = 0; c < nchunks; ++c) {
    if (c + 1 < nchunks) {
      issue((c + 1) * S1_KCH, (c + 1) & 1);
      asm volatile("s_wait_asynccnt 2" ::: "memory");
    } else {
      asm volatile("s_wait_asynccnt 0" ::: "memory");
    }
    __syncthreads();

    const int kn  = min(S1_KCH, nlon - c * S1_KCH);   // multiple of 4
    const int buf = c & 1;
    if (active) {
      for (int k = 0; k < kn; k += 4) {
        // A fragment (16x4 f32): VGPR0 = K{0,2}, VGPR1 = K{1,3}
        v2f a = *(const v2f*)(Wbase + c * S1_KCH + k);
        #pragma unroll
        for (int b2 = 0; b2 < 2; ++b2) {
          #pragma unroll
          for (int vt = 0; vt < 5; ++vt) {
            v2f bf;
            bf.x = Xs[buf][b2][k + 2 * lhi]    [vt * 16 + lrow];
            bf.y = Xs[buf][b2][k + 2 * lhi + 1][vt * 16 + lrow];
            acc[b2][vt] = __builtin_amdgcn_wmma_f32_16x16x4_f32(
                false, a, false, bf, (short)0, acc[b2][vt], false, false);
          }
        }
      }
    }
    __syncthreads();
  }

  // store F[b][r][ri][m][v]; zero-fill tiles beyond spectral truncation
  #pragma unroll
  for (int b2 = 0; b2 < 2; ++b2) {
    float* Fb = F + ((((size_t)b2 * NRINGS_ + r) * 2 + ri) * MMAX_) * NVAR_;
    #pragma unroll
    for (int vt = 0; vt < 5; ++vt) {
      #pragma unroll
      for (int j = 0; j < 8; ++j) {
        const int mrow = m0 + j + 8 * lhi;
        Fb[(size_t)mrow * NVAR_ + vt * 16 + lrow] =
            active ? acc[b2][vt][j] : 0.0f;
      }
    }
  }
}

// ---------------------------------------------------------------------------
// Stage 2: per (m, batch) block: fold NH/SH rings into sym/anti panels in LDS,
// then 4 GEMMs (80x160x80): odd = P_s * sym, even = P_a * anti, re & im.
// 320 threads = 10 waves: wave = (pp in {sym,anti}) x (ltile in 0..4); each
// wave holds 10 accumulators (2 ri x 5 v-tiles) sharing one A fragment.
// ---------------------------------------------------------------------------
#define S2_ICH 20    // i-rows per chunk: 2*2*20*80*4 = 25600 B

__global__ __launch_bounds__(320) void sht_leg_stage2(
    const float* __restrict__ F, const float* __restrict__ Ps,
    const float* __restrict__ Pa, float* __restrict__ out)
{
  __shared__ float SA[2][2][S2_ICH][NVAR_];  // [sym/anti][re/im][i][v]

  const int m = blockIdx.x;
  const int b = blockIdx.y;

  const int tid  = threadIdx.x;
  const int wave = tid >> 5;
  const int lane = tid & 31;
  const int lrow = lane & 15;
  const int lhi  = lane >> 4;

  const int pp = wave / 5;   // 0: symmetric -> odd rows, 1: antisymmetric -> even
  const int lt = wave % 5;
  const int l0 = lt * 16;

  const float* P = (pp == 0 ? Ps : Pa) +
      ((size_t)m * 80 + (size_t)(l0 + lrow)) * NLAT_ + 2 * lhi;

  const float* Fb = F + (size_t)b * NRINGS_ * 2 * MMAX_ * NVAR_ +
                    (size_t)m * NVAR_;

  v8f acc[2][5] = {};

  for (int ic = 0; ic < NLAT_; ic += S2_ICH) {
    // fold chunk: S = nh + sh, A = nh - sh  (sh ring = 319 - i)
    for (int idx = tid; idx < 2 * S2_ICH * (NVAR_ / 4); idx += 320) {
      const int f   = idx % (NVAR_ / 4);
      const int il  = (idx / (NVAR_ / 4)) % S2_ICH;
      const int riL = idx / ((NVAR_ / 4) * S2_ICH);
      const int i   = ic + il;
      const float4 nh = *(const float4*)(
          Fb + (size_t)(i * 2 + riL) * MMAX_ * NVAR_ + f * 4);
      const float4 sh = *(const float4*)(
          Fb + (size_t)((NRINGS_ - 1 - i) * 2 + riL) * MMAX_ * NVAR_ + f * 4);
      float4 s, d;
      s.x = nh.x + sh.x; s.y = nh.y + sh.y; s.z = nh.z + sh.z; s.w = nh.w + sh.w;
      d.x = nh.x - sh.x; d.y = nh.y - sh.y; d.z = nh.z - sh.z; d.w = nh.w - sh.w;
      *(float4*)(&SA[0][riL][il][f * 4]) = s;
      *(float4*)(&SA[1][riL][il][f * 4]) = d;
    }
    __syncthreads();

    for (int k = 0; k < S2_ICH; k += 4) {
      v2f a = *(const v2f*)(P + ic + k);
      #pragma unroll
      for (int vt = 0; vt < 5; ++vt) {
        #pragma unroll
        for (int riL = 0; riL < 2; ++riL) {
          v2f bf;
          bf.x = SA[pp][riL][k + 2 * lhi]    [vt * 16 + lrow];
          bf.y = SA[pp][riL][k + 2 * lhi + 1][vt * 16 + lrow];
          acc[riL][vt] = __builtin_amdgcn_wmma_f32_16x16x4_f32(
              false, a, false, bf, (short)0, acc[riL][vt], false, false);
        }
      }
    }
    __syncthreads();
  }

  // store complex64 output: spec[b][e][2l+parity][m][v] (re,im interleaved)
  const int parity = (pp == 0) ? 1 : 0;   // symmetric -> odd rows
  #pragma unroll
  for (int vt = 0; vt < 5; ++vt) {
    #pragma unroll
    for (int j = 0; j < 8; ++j) {
      const int row = 2 * (l0 + j + 8 * lhi) + parity;
      const size_t off =
          ((((size_t)b * MMAX_ + row) * MMAX_ + m) * NVAR_ + vt * 16 + lrow) * 2;
      float2 c;
      c.x = acc[0][vt][j];
      c.y = acc[1][vt][j];
      *(float2*)(out + off) = c;
    }
  }
}

// ---------------------------------------------------------------------------
extern "C" void kernel_launch(void* const* d_in, const int* in_sizes, int n_in,
                              void* d_out, int out_size, void* d_ws, size_t ws_size,
                              hipStream_t stream) {
  (void)in_sizes; (void)n_in; (void)out_size; (void)ws_size;
  const float* x  = (const float*)d_in[0];   // (B,E,NPOINTS,NVAR)
  const float* Ps = (const float*)d_in[1];   // (MMAX,80,NLAT) symmetric
  const float* Pa = (const float*)d_in[2];   // (MMAX,80,NLAT) antisymmetric
  const float* Wr = (const float*)d_in[3];   // (NRINGS,MMAX,MAXLON)
  const float* Wi = (const float*)d_in[4];

  float* F   = (float*)d_ws;   // 2*320*2*160*80 f32 = 65.5 MB
  float* out = (float*)d_out;  // complex64 viewed as interleaved f32

  sht_dft_stage1<<<dim3(NRINGS_), 640, 0, stream>>>(x, Wr, Wi, F);
  sht_leg_stage2<<<dim3(MMAX_, NB_), 320, 0, stream>>>(F, Ps, Pa, out);
}